// GCN_55800215109855
// MI455X (gfx1250) — compile-verified
//
#include <hip/hip_runtime.h>

typedef __attribute__((ext_vector_type(2))) float v2f;
typedef __attribute__((ext_vector_type(8))) float v8f;

static constexpr int   HD    = 128;   // hidden dim (= F_IN)
static constexpr int   OUTD  = 64;    // output dim
static constexpr float ALPHA = 0.25f;
static constexpr float EPSV  = 1e-5f;

// Native CDNA5 fp32 global atomic add (no CAS loop), device scope.
__device__ __forceinline__ void atomAddF(float* p, float v) {
    asm volatile("global_atomic_add_f32 %0, %1, off scope:SCOPE_DEV"
                 : : "v"(p), "v"(v) : "memory");
}
__device__ __forceinline__ void atomAdd4F(float* p, float4 v) {
    asm volatile("global_atomic_add_f32 %0, %1, off scope:SCOPE_DEV\n\t"
                 "global_atomic_add_f32 %0, %2, off offset:4 scope:SCOPE_DEV\n\t"
                 "global_atomic_add_f32 %0, %3, off offset:8 scope:SCOPE_DEV\n\t"
                 "global_atomic_add_f32 %0, %4, off offset:12 scope:SCOPE_DEV"
                 : : "v"(p), "v"(v.x), "v"(v.y), "v"(v.z), "v"(v.w) : "memory");
}

// ---------------------------------------------------------------- utilities
__global__ void fill_f32(float* __restrict__ p, float v, int n) {
    int i = blockIdx.x * blockDim.x + threadIdx.x;
    if (i < n) p[i] = v;
}

// deg[dst[e]] += edge_attr[e,3]   (deg pre-filled with 1.0 = self loop weight)
__global__ void deg_accum(const int* __restrict__ dst, const float* __restrict__ ea,
                          float* __restrict__ deg, int E) {
    int e = blockIdx.x * blockDim.x + threadIdx.x;
    if (e < E) atomAddF(&deg[dst[e]], ea[e * 4 + 3]);
}

__global__ void dinv_kernel(const float* __restrict__ deg, float* __restrict__ dinv, int n) {
    int i = blockIdx.x * blockDim.x + threadIdx.x;
    if (i < n) {
        float d = deg[i];
        dinv[i] = d > 0.f ? rsqrtf(fmaxf(d, 1e-30f)) : 0.f;
    }
}

__global__ void norm_kernel(const int* __restrict__ src, const int* __restrict__ dst,
                            const float* __restrict__ ea, const float* __restrict__ dinv,
                            float* __restrict__ nrm, int E) {
    int e = blockIdx.x * blockDim.x + threadIdx.x;
    if (e < E) nrm[e] = dinv[src[e]] * ea[e * 4 + 3] * dinv[dst[e]];
}

// m = ALPHA*x0 + (1-ALPHA)*dinv[n]^2 * h   (self-loop term folded in), float4 wide
__global__ void init_m(const float4* __restrict__ x0, const float4* __restrict__ h,
                       const float* __restrict__ dinv, float4* __restrict__ m, int total4) {
    int i = blockIdx.x * blockDim.x + threadIdx.x;
    if (i < total4) {
        int n = i >> 5;                       // 32 float4 per row of 128
        float sn = dinv[n];
        float c = (1.f - ALPHA) * sn * sn;
        float4 a = x0[i], b = h[i], o;
        o.x = ALPHA * a.x + c * b.x;
        o.y = ALPHA * a.y + c * b.y;
        o.z = ALPHA * a.z + c * b.z;
        o.w = ALPHA * a.w + c * b.w;
        m[i] = o;
    }
}

// one wave (32 lanes) per edge; float4 per lane; native f32 atomics
__global__ void scatter_edges(const int* __restrict__ src, const int* __restrict__ dst,
                              const float* __restrict__ nrm, const float* __restrict__ h,
                              float* __restrict__ m, int E) {
    int e = blockIdx.x * (blockDim.x >> 5) + (threadIdx.x >> 5);
    if (e >= E) return;
    int lane = threadIdx.x & 31;
    int s = src[e], d = dst[e];
    float coef = (1.f - ALPHA) * nrm[e];
    float4 v = ((const float4*)(h + (size_t)s * HD))[lane];
    v.x *= coef; v.y *= coef; v.z *= coef; v.w *= coef;
    atomAdd4F(m + (size_t)d * HD + lane * 4, v);
}

// ---------------------------------------------------------------- WMMA GEMM
// C[M x NCOLS] = A[M x 128] * B[128 x NCOLS] + bias
// TRANSB=0 -> B[k][j] = W[k*NCOLS + j]   (W_conv, (H,H))
// TRANSB=1 -> B[k][j] = W[j*128 + k]     (W_lin1/W_lin2, (cols,128))
// W staged ONCE per block, k-pair-packed so each lane's B operand is one ds_load_b64:
//   wlds[(k>>1)*BSTR + j*2 + (k&1)]   (BSTR padded so lane halves hit disjoint banks)
// Each block processes MT M-tiles of 16 rows; wave w owns columns [16w,16w+16).
template <int NCOLS, bool TRANSB, int MT>
__global__ __launch_bounds__(32 * (NCOLS / 16)) void gemm_wmma(
    const float* __restrict__ A, const float* __restrict__ W,
    const float* __restrict__ bias, float* __restrict__ C, int M) {
    constexpr int BSTR = NCOLS * 2 + 32;
    __shared__ float wlds[(HD / 2) * BSTR];
    __shared__ float alds[16 * 132];          // padded stride (bank-conflict free)

    const int tid  = threadIdx.x;
    const int lane = tid & 31;
    const int wv   = tid >> 5;
    const int n0   = wv * 16;
    const int cl   = lane & 15;               // A-row / B,C-col within tile
    const int koff = (lane >> 4) << 1;        // lanes 16-31 hold K+2,K+3

    // stage W once (pair-packed; linear coalesced reads)
    for (int i = tid; i < NCOLS * HD; i += blockDim.x) {
        int k, j;
        if (TRANSB) { j = i / HD;    k = i - j * HD;    }
        else        { k = i / NCOLS; j = i - k * NCOLS; }
        wlds[(k >> 1) * BSTR + j * 2 + (k & 1)] = W[i];
    }

    const float bv = bias ? bias[n0 + cl] : 0.f;

    for (int t = 0; t < MT; ++t) {
        const int m0 = (blockIdx.x * MT + t) * 16;
        if (m0 >= M) break;                   // uniform branch
        __syncthreads();                      // W ready (t=0) / alds free (t>0)
        for (int i = tid; i < 16 * HD; i += blockDim.x) {
            int r = i >> 7, k = i & (HD - 1);
            alds[r * 132 + k] = A[(size_t)(m0 + r) * HD + k];
        }
        __syncthreads();

        v8f c;
#pragma unroll
        for (int r = 0; r < 8; ++r) c[r] = bv;

#pragma unroll
        for (int k = 0; k < HD; k += 4) {
            // A 16x4: lane L<16 -> {A[L][k],A[L][k+1]}, L>=16 -> {A[L-16][k+2],A[L-16][k+3]}
            v2f a = *(const v2f*)&alds[cl * 132 + k + koff];
            // B 4x16: contiguous pair {B[k+koff][n0+cl], B[k+koff+1][n0+cl]}
            v2f b = *(const v2f*)&wlds[((k + koff) >> 1) * BSTR + (n0 + cl) * 2];
            c = __builtin_amdgcn_wmma_f32_16x16x4_f32(false, a, false, b, (short)0, c,
                                                      false, false);
        }

        // C 16x16: VGPR r -> row m0+r (lanes 0-15) / m0+r+8 (lanes 16-31)
        const int rbase = m0 + ((lane >> 4) << 3);
#pragma unroll
        for (int r = 0; r < 8; ++r)
            C[(size_t)(rbase + r) * NCOLS + n0 + cl] = c[r];
    }
}

// ---------------------------------------------------------------- batchnorm
__global__ void bn_stats(const float* __restrict__ g, float* __restrict__ sums, int M) {
    int col = threadIdx.x;                    // HD threads, column-per-thread (coalesced)
    float s = 0.f, s2 = 0.f;
    for (int r = blockIdx.x; r < M; r += gridDim.x) {
        float v = g[(size_t)r * HD + col];
        s += v;
        s2 += v * v;
    }
    atomAddF(&sums[col], s);
    atomAddF(&sums[HD + col], s2);
}

__global__ void bn_finalize(const float* __restrict__ sums, const float* __restrict__ gamma,
                            const float* __restrict__ beta, float* __restrict__ ss, int M) {
    int c = threadIdx.x;
    if (c < HD) {
        float inv = 1.f / (float)M;
        float mu  = sums[c] * inv;
        float var = sums[HD + c] * inv - mu * mu;
        float sc  = gamma[c] * rsqrtf(var + EPSV);
        ss[c]      = sc;
        ss[HD + c] = beta[c] - mu * sc;
    }
}

__global__ void bn_apply_relu(const float4* __restrict__ g, const float* __restrict__ ss,
                              float4* __restrict__ h, int total4) {
    int i = blockIdx.x * blockDim.x + threadIdx.x;
    if (i < total4) {
        int c0 = (i << 2) & (HD - 1);
        float4 sc = *(const float4*)&ss[c0];
        float4 sh = *(const float4*)&ss[HD + c0];
        float4 v = g[i], o;
        o.x = fmaxf(0.f, v.x * sc.x + sh.x);
        o.y = fmaxf(0.f, v.y * sc.y + sh.y);
        o.z = fmaxf(0.f, v.z * sc.z + sh.z);
        o.w = fmaxf(0.f, v.w * sc.w + sh.w);
        h[i] = o;
    }
}

// ---------------------------------------------------------------- driver
extern "C" void kernel_launch(void* const* d_in, const int* in_sizes, int n_in,
                              void* d_out, int out_size, void* d_ws, size_t ws_size,
                              hipStream_t stream) {
    const float* x     = (const float*)d_in[0];
    const int*   ei    = (const int*)d_in[1];
    const float* ea    = (const float*)d_in[2];
    const float* Wl1   = (const float*)d_in[3];
    const float* bl1   = (const float*)d_in[4];
    const float* Wc1   = (const float*)d_in[5];
    const float* Wc2   = (const float*)d_in[6];
    const float* Wc3   = (const float*)d_in[7];
    const float* gamma = (const float*)d_in[8];
    const float* beta  = (const float*)d_in[9];
    const float* Wl2   = (const float*)d_in[10];
    const float* bl2   = (const float*)d_in[11];

    const int N = in_sizes[0] / HD;
    const int E = in_sizes[1] / 2;
    const int* src = ei;
    const int* dst = ei + E;

    float* ws   = (float*)d_ws;
    size_t NH   = (size_t)N * HD;
    float* x0   = ws;
    float* buf1 = x0 + NH;
    float* buf2 = buf1 + NH;
    float* deg  = buf2 + NH;
    float* dinv = deg + N;
    float* nrm  = dinv + N;
    float* sums = nrm + E;
    float* ss   = sums + 2 * HD;

    float* hout = (float*)d_out;       // h3: N x 128
    float* oout = hout + NH;           // o : N x 64

    int total4 = (int)(NH >> 2);
    int gElem4 = (total4 + 255) / 256;
    int gE     = (E + 255) / 256;
    int gN     = (N + 255) / 256;
    constexpr int MT = 8;              // M-tiles per block (W staged once per 128 rows)
    int gGemm  = (N + 16 * MT - 1) / (16 * MT);

    // degree / normalization coefficients
    fill_f32<<<gN, 256, 0, stream>>>(deg, 1.0f, N);
    deg_accum<<<gE, 256, 0, stream>>>(dst, ea, deg, E);
    dinv_kernel<<<gN, 256, 0, stream>>>(deg, dinv, N);
    norm_kernel<<<gE, 256, 0, stream>>>(src, dst, ea, dinv, nrm, E);

    // x0 = x @ W_lin1^T + b_lin1
    gemm_wmma<HD, true, MT><<<gGemm, 256, 0, stream>>>(x, Wl1, bl1, x0, N);

    struct Layer { const float* W; float* m; float* g; float* h; const float* hin; };
    Layer L[3] = {
        {Wc1, buf1, buf2, buf1, x0},    // h1 = buf1
        {Wc2, buf2, buf1, buf2, buf1},  // h2 = buf2
        {Wc3, buf1, buf2, hout, buf2},  // h3 = d_out
    };
    for (int l = 0; l < 3; ++l) {
        init_m<<<gElem4, 256, 0, stream>>>((const float4*)x0, (const float4*)L[l].hin,
                                           dinv, (float4*)L[l].m, total4);
        scatter_edges<<<(E + 7) / 8, 256, 0, stream>>>(src, dst, nrm, L[l].hin, L[l].m, E);
        gemm_wmma<HD, false, MT><<<gGemm, 256, 0, stream>>>(L[l].m, L[l].W, nullptr,
                                                            L[l].g, N);
        fill_f32<<<1, 256, 0, stream>>>(sums, 0.f, 2 * HD);
        bn_stats<<<512, HD, 0, stream>>>(L[l].g, sums, N);
        bn_finalize<<<1, HD, 0, stream>>>(sums, gamma, beta, ss, N);
        bn_apply_relu<<<gElem4, 256, 0, stream>>>((const float4*)L[l].g, ss,
                                                  (float4*)L[l].h, total4);
    }

    // o = h3 @ W_lin2^T + b_lin2
    gemm_wmma<OUTD, true, MT><<<gGemm, 128, 0, stream>>>(hout, Wl2, bl2, oout, N);
}